// S4Model_83975200571707
// MI455X (gfx1250) — compile-verified
//
#include <hip/hip_runtime.h>
#include <math.h>

// S4D forward for MI455X (gfx1250), f32 WMMA 16x16x4 path + async global->LDS.
// Chunked state-space reformulation of the FFT causal conv:
//   per (b,h): S = A1@U ; scan over 64 chunks ; Y = T@U + G@sprev
// All GEMMs run on V_WMMA_F32_16X16X4_F32 (exact f32, matches reference math).
// LDS fills use GLOBAL_LOAD_ASYNC_TO_LDS_B32/B128 (ASYNCcnt) per CDNA5 ISA ch.8.
// Workspace requirement: ~74 MB (h + y + per-layer matrices).

typedef __attribute__((ext_vector_type(2))) float v2f;
typedef __attribute__((ext_vector_type(8))) float v8f;

#define S4_B 16
#define S4_L 4096
#define S4_H 128
#define S4_N 32
#define S4_Q 64

__device__ __forceinline__ v8f wmma_f32(v2f a, v2f b, v8f c) {
  // 8 args: (neg_a, A, neg_b, B, c_mod, C, reuse_a, reuse_b)
  return __builtin_amdgcn_wmma_f32_16x16x4_f32(false, a, false, b, (short)0, c,
                                               false, false);
}

// ---- async global->LDS copies (CDNA5, tracked with ASYNCcnt) --------------
// Generic shared pointers carry the LDS byte offset in their low 32 bits
// (ISA 10.2: LDS aperture address is truncated to addr[31:0]).
__device__ __forceinline__ unsigned lds_off(const void* p) {
  return (unsigned)(size_t)p;
}
__device__ __forceinline__ void async_b32(unsigned lds, const void* g) {
  asm volatile("global_load_async_to_lds_b32 %0, %1, off"
               :: "v"(lds), "v"(g) : "memory");
}
__device__ __forceinline__ void async_b128(unsigned lds, const void* g) {
  asm volatile("global_load_async_to_lds_b128 %0, %1, off"
               :: "v"(lds), "v"(g) : "memory");
}
__device__ __forceinline__ void wait_async0() {
  asm volatile("s_wait_asynccnt 0" ::: "memory");
}

// Fragment load for f32 WMMA. With A stored K-major ([k][m]) and B stored
// row-major ([k][n]), both operands use the identical access pattern:
//   lanes 0-15 hold K=0,1 for x=lane; lanes 16-31 hold K=2,3.
__device__ __forceinline__ v2f frag_ld(const float* buf, int k0, int x0, int ld,
                                       int lane) {
  int x = lane & 15;
  int kk = (lane >> 4) << 1;
  v2f f;
  f.x = buf[(k0 + kk) * ld + x0 + x];
  f.y = buf[(k0 + kk + 1) * ld + x0 + x];
  return f;
}

// C/D tile store: VGPR r holds M=r (lanes 0-15) and M=r+8 (lanes 16-31), N=lane&15.
__device__ __forceinline__ void tile_st(float* buf, v8f acc, int m0, int n0,
                                        int ld, int lane) {
  int n = lane & 15;
  int mh = (lane >> 4) << 3;
#pragma unroll
  for (int r = 0; r < 8; ++r) buf[(m0 + mh + r) * ld + n0 + n] = acc[r];
}

__device__ __forceinline__ float gelu_tanh(float x) {
  const float c0 = 0.7978845608028654f;  // sqrt(2/pi)
  float x3 = x * x * x;
  return 0.5f * x * (1.0f + tanhf(c0 * (x + 0.044715f * x3)));
}

// ---------------------------------------------------------------------------
// Embed: h[b,hh,l] = sum_d x[b,l,d]*ew[d,hh] + eb[hh]   (K=32, bandwidth-bound)
// grid = B * (L/64), block = 256
// ---------------------------------------------------------------------------
__global__ __launch_bounds__(256) void s4_embed_kernel(
    const float* __restrict__ x, const float* __restrict__ ew,
    const float* __restrict__ eb, float* __restrict__ hout) {
  __shared__ __align__(16) float sX[64 * 32];
  int blk = blockIdx.x;
  int b = blk >> 6;
  int l0 = (blk & 63) * 64;
  int t = threadIdx.x;
  // x tile is contiguous: one b128 per thread x2
  const float* xs = x + ((size_t)b * S4_L + l0) * 32;
  async_b128(lds_off(&sX[t * 4]), xs + t * 4);
  async_b128(lds_off(&sX[t * 4 + 1024]), xs + t * 4 + 1024);
  wait_async0();
  __syncthreads();
  for (int idx = t; idx < S4_H * 64; idx += 256) {
    int hh = idx >> 6, c = idx & 63;
    float s = eb[hh];
#pragma unroll
    for (int d = 0; d < 32; ++d) s += sX[c * 32 + d] * ew[d * S4_H + hh];
    hout[((size_t)b * S4_H + hh) * S4_L + l0 + c] = s;
  }
}

// ---------------------------------------------------------------------------
// Per-layer prep: builds the chunk matrices (all K-major for WMMA A-operand):
//  A1t[h][j][m]  m<32: Re(w_m^(63-j)), m>=32: Im(w_(m-32)^(63-j))
//  Tt [h][j][i]  = K[i-j] for i>=j else 0,  K[t] = 2 Re sum_n Ck_n w_n^t
//  Gt [h][n][i]  n<32:  2 Re(Ck_n w_n^(i+1)),  n>=32: -2 Im(Ck w^(i+1))
//  wq [h][0..31]=Re(w^64), [32..63]=Im(w^64)
//  Wt [h][g]    = out_w[g][h]   (transposed GLU weight, K-major)
// grid = 128 (one block per h), block = 64
// ---------------------------------------------------------------------------
__global__ __launch_bounds__(64) void s4_prep_kernel(
    const float* __restrict__ log_dt, const float* __restrict__ Are,
    const float* __restrict__ Aim, const float* __restrict__ Cre,
    const float* __restrict__ Cim, const float* __restrict__ outw,
    float* __restrict__ A1t, float* __restrict__ Tt, float* __restrict__ Gt,
    float* __restrict__ wq, float* __restrict__ Wt) {
  __shared__ float dAr[32], dAi[32], Ckr[32], Cki[32], Kk[64];
  int h = blockIdx.x;
  int t = threadIdx.x;
  if (t < 32) {
    int n = t;
    float dt = expf(log_dt[h]);
    float ar = Are[h * 32 + n], ai = Aim[h * 32 + n];
    float dr = dt * ar, di = dt * ai;
    dAr[n] = dr;
    dAi[n] = di;
    float er = expf(dr);
    float wr_ = er * cosf(di), wi_ = er * sinf(di);
    // Ck = C * (w - 1) / A
    float nr = wr_ - 1.f, ni = wi_;
    float inv = 1.f / (ar * ar + ai * ai);
    float tr_ = (nr * ar + ni * ai) * inv;
    float ti_ = (ni * ar - nr * ai) * inv;
    float cr = Cre[h * 32 + n], ci = Cim[h * 32 + n];
    Ckr[n] = cr * tr_ - ci * ti_;
    Cki[n] = cr * ti_ + ci * tr_;
    // w^Q
    float eq = expf(64.f * dr);
    wq[h * 64 + n] = eq * cosf(64.f * di);
    wq[h * 64 + 32 + n] = eq * sinf(64.f * di);
  }
  __syncthreads();
  if (t < 64) {
    float s = 0.f;
    for (int n = 0; n < 32; ++n) {
      float e = expf((float)t * dAr[n]);
      float er = e * cosf((float)t * dAi[n]);
      float ei = e * sinf((float)t * dAi[n]);
      s += 2.f * (Ckr[n] * er - Cki[n] * ei);
    }
    Kk[t] = s;
  }
  __syncthreads();
  for (int idx = t; idx < 4096; idx += 64) {
    int j = idx >> 6, row = idx & 63;
    // A1t
    {
      int n = row & 31;
      float p = (float)(63 - j);
      float e = expf(p * dAr[n]);
      A1t[h * 4096 + idx] =
          (row < 32) ? e * cosf(p * dAi[n]) : e * sinf(p * dAi[n]);
    }
    // Tt: (j=k, row=i)
    Tt[h * 4096 + idx] = (row >= j) ? Kk[row - j] : 0.f;
    // Gt: (j=n2, row=i)
    {
      int nn = j & 31;
      float p2 = (float)(row + 1);
      float e2 = expf(p2 * dAr[nn]);
      float er2 = e2 * cosf(p2 * dAi[nn]);
      float ei2 = e2 * sinf(p2 * dAi[nn]);
      float cwr = Ckr[nn] * er2 - Cki[nn] * ei2;
      float cwi = Ckr[nn] * ei2 + Cki[nn] * er2;
      Gt[h * 4096 + idx] = (j < 32) ? 2.f * cwr : -2.f * cwi;
    }
  }
  for (int g = t; g < 256; g += 64) Wt[h * 256 + g] = outw[g * S4_H + h];
}

// ---------------------------------------------------------------------------
// S4D conv per (b,h) channel: chunked scan, three 64x64x64 f32-WMMA GEMMs,
// fused D_skip + gelu epilogue.  grid = B*H = 2048, block = 256 (8 waves).
// All LDS fills issued asynchronously up front; A1/T/G stay resident (gfx1250
// has 320 KB LDS/WGP, ~80 KB here -> 3 blocks/WGP), removing two barrier
// phases vs. a single reused A-buffer.
// ---------------------------------------------------------------------------
__global__ __launch_bounds__(256) void s4_conv_kernel(
    const float* __restrict__ hin, const float* __restrict__ A1t,
    const float* __restrict__ Tt, const float* __restrict__ Gt,
    const float* __restrict__ wq, const float* __restrict__ Dskip,
    float* __restrict__ yout) {
  __shared__ __align__(16) float sU[64 * 65];  // U[j][c]=u[c*64+j], pad 65
  __shared__ __align__(16) float sS[64 * 65];  // S -> sprev -> Y
  __shared__ __align__(16) float sA1[4096];    // K-major A operands
  __shared__ __align__(16) float sT[4096];
  __shared__ __align__(16) float sG[4096];
  int blk = blockIdx.x;
  int h = blk & (S4_H - 1);
  int t = threadIdx.x;
  int wid = t >> 5, lane = t & 31;
  const float* u = hin + (size_t)blk * S4_L;

  // ---- async fills: U (per-lane LDS scatter, b32) + matrices (b128) ----
  for (int l = t; l < S4_L; l += 256)
    async_b32(lds_off(&sU[(l & 63) * 65 + (l >> 6)]), u + l);
  for (int i = t * 4; i < 4096; i += 1024) {
    async_b128(lds_off(&sA1[i]), A1t + h * 4096 + i);
    async_b128(lds_off(&sT[i]), Tt + h * 4096 + i);
    async_b128(lds_off(&sG[i]), Gt + h * 4096 + i);
  }
  wait_async0();
  __syncthreads();

  // ---- S = A1 @ U ----
  int t0 = wid * 2;
  int m0a = (t0 >> 2) * 16, n0a = (t0 & 3) * 16;
  int m0b = ((t0 + 1) >> 2) * 16, n0b = ((t0 + 1) & 3) * 16;
  {
    v8f a0 = {}, a1 = {};
#pragma unroll
    for (int k = 0; k < 64; k += 4) {
      a0 = wmma_f32(frag_ld(sA1, k, m0a, 64, lane),
                    frag_ld(sU, k, n0a, 65, lane), a0);
      a1 = wmma_f32(frag_ld(sA1, k, m0b, 64, lane),
                    frag_ld(sU, k, n0b, 65, lane), a1);
    }
    tile_st(sS, a0, m0a, n0a, 65, lane);
    tile_st(sS, a1, m0b, n0b, 65, lane);
  }
  __syncthreads();

  // ---- exclusive scan over chunks (wave 0), S -> sprev in place ----
  if (wid == 0) {
    float wr = wq[h * 64 + lane];
    float wi = wq[h * 64 + 32 + lane];
    float sr = 0.f, si = 0.f;
    for (int c = 0; c < 64; ++c) {
      float Sre = sS[lane * 65 + c];
      float Sim = sS[(lane + 32) * 65 + c];
      sS[lane * 65 + c] = sr;
      sS[(lane + 32) * 65 + c] = si;
      float nsr = wr * sr - wi * si + Sre;
      float nsi = wr * si + wi * sr + Sim;
      sr = nsr;
      si = nsi;
    }
  }
  __syncthreads();

  // ---- Y = T @ U + G @ sprev (single fused accumulation chain) ----
  v8f y0 = {}, y1 = {};
#pragma unroll
  for (int k = 0; k < 64; k += 4) {
    y0 = wmma_f32(frag_ld(sT, k, m0a, 64, lane),
                  frag_ld(sU, k, n0a, 65, lane), y0);
    y1 = wmma_f32(frag_ld(sT, k, m0b, 64, lane),
                  frag_ld(sU, k, n0b, 65, lane), y1);
    y0 = wmma_f32(frag_ld(sG, k, m0a, 64, lane),
                  frag_ld(sS, k, n0a, 65, lane), y0);
    y1 = wmma_f32(frag_ld(sG, k, m0b, 64, lane),
                  frag_ld(sS, k, n0b, 65, lane), y1);
  }
  __syncthreads();
  tile_st(sS, y0, m0a, n0a, 65, lane);  // overwrite sprev with Y
  tile_st(sS, y1, m0b, n0b, 65, lane);
  __syncthreads();

  // ---- epilogue: y = gelu(conv + D_skip*u), coalesced store ----
  float dsk = Dskip[h];
  float* yo = yout + (size_t)blk * S4_L;
  for (int l = t; l < S4_L; l += 256) {
    float uv = sU[(l & 63) * 65 + (l >> 6)];
    float yv = sS[(l & 63) * 65 + (l >> 6)] + dsk * uv;
    yo[l] = gelu_tanh(yv);
  }
}

// ---------------------------------------------------------------------------
// GLU + residual + channel LayerNorm, fused behind a 256x64x128 WMMA GEMM.
// V = Wt^T-form GEMM over the h-channel (K=128), then z = V[:128]*sig(V[128:])
// + h_old, then LN over the 128 channels per position.  In-place on h.
// grid = B * (L/64) = 1024, block = 256 (8 waves, 8 tiles each).
// ---------------------------------------------------------------------------
__global__ __launch_bounds__(256) void s4_glu_ln_kernel(
    const float* __restrict__ y, const float* __restrict__ Wt,
    const float* __restrict__ outb, float* __restrict__ hio,
    const float* __restrict__ nw, const float* __restrict__ nb) {
  __shared__ __align__(16) float sV[256 * 64];  // 64 KB; K-slabs alias front
  float* sW = sV;                               // 16 x 256 (K-major W slab)
  float* sY = sV + 16 * 256;                    // 16 x 64  (Y slab)
  int blk = blockIdx.x;
  int b = blk >> 6;
  int l0 = (blk & 63) * 64;
  int t = threadIdx.x;
  int wid = t >> 5, lane = t & 31;
  const float* yb = y + (size_t)b * S4_H * S4_L;
  float* hb = hio + (size_t)b * S4_H * S4_L;

  v8f acc[8];
#pragma unroll
  for (int i = 0; i < 8; ++i) acc[i] = {};
  int tbase = wid * 8;

  for (int k0 = 0; k0 < S4_H; k0 += 16) {
    __syncthreads();
    {
      // W slab: 16 contiguous floats per thread (4x b128)
      const float* wsrc = Wt + k0 * 256 + t * 16;
      unsigned wdst = lds_off(&sW[t * 16]);
      async_b128(wdst, wsrc);
      async_b128(wdst + 16, wsrc + 4);
      async_b128(wdst + 32, wsrc + 8);
      async_b128(wdst + 48, wsrc + 12);
      // Y slab: one b128 per thread, row = t>>4, col = (t&15)*4
      int row = t >> 4, col = (t & 15) * 4;
      async_b128(lds_off(&sY[row * 64 + col]),
                 yb + (size_t)(k0 + row) * S4_L + l0 + col);
    }
    wait_async0();
    __syncthreads();
#pragma unroll
    for (int i = 0; i < 8; ++i) {
      int ti = tbase + i;
      int m0 = (ti >> 2) * 16, n0 = (ti & 3) * 16;
#pragma unroll
      for (int k = 0; k < 16; k += 4) {
        acc[i] = wmma_f32(frag_ld(sW, k, m0, 256, lane),
                          frag_ld(sY, k, n0, 64, lane), acc[i]);
      }
    }
  }
  __syncthreads();
  // V tiles -> LDS with bias
#pragma unroll
  for (int i = 0; i < 8; ++i) {
    int ti = tbase + i;
    int m0 = (ti >> 2) * 16, n0 = (ti & 3) * 16;
    int n = lane & 15, mh = (lane >> 4) << 3;
#pragma unroll
    for (int r = 0; r < 8; ++r) {
      int g = m0 + mh + r;
      sV[g * 64 + n0 + n] = acc[i][r] + outb[g];
    }
  }
  __syncthreads();
  // GLU + residual + LN: 64 threads, one column (position) each
  if (t < 64) {
    int c = t;
    int l = l0 + c;
    float mu = 0.f;
    for (int hh = 0; hh < S4_H; ++hh) {
      float a = sV[hh * 64 + c];
      float gt = sV[(hh + S4_H) * 64 + c];
      float z = a * (1.f / (1.f + expf(-gt))) + hb[(size_t)hh * S4_L + l];
      sV[hh * 64 + c] = z;
      mu += z;
    }
    mu *= (1.f / 128.f);
    float var = 0.f;
    for (int hh = 0; hh < S4_H; ++hh) {
      float d = sV[hh * 64 + c] - mu;
      var += d * d;
    }
    var *= (1.f / 128.f);
    float rs = rsqrtf(var + 1e-5f);
    for (int hh = 0; hh < S4_H; ++hh)
      hb[(size_t)hh * S4_L + l] = (sV[hh * 64 + c] - mu) * rs * nw[hh] + nb[hh];
  }
}

// ---------------------------------------------------------------------------
// Decode: out[b,l,o] = sum_hh h[b,hh,l]*dec_w[hh,o] + dec_b[o]
// grid = B * (L/64) = 1024, block = 256
// ---------------------------------------------------------------------------
__global__ __launch_bounds__(256) void s4_decode_kernel(
    const float* __restrict__ hin, const float* __restrict__ dw,
    const float* __restrict__ db, float* __restrict__ out) {
  __shared__ __align__(16) float sH[128 * 64];
  int blk = blockIdx.x;
  int b = blk >> 6;
  int l0 = (blk & 63) * 64;
  int t = threadIdx.x;
  // 32 floats per thread, rows of 64: row = (i>>6), col = i&63; use b128s
  for (int i4 = t * 4; i4 < 128 * 64; i4 += 1024) {
    int hh = i4 >> 6, c = i4 & 63;
    async_b128(lds_off(&sH[i4]),
               hin + ((size_t)b * S4_H + hh) * S4_L + l0 + c);
  }
  wait_async0();
  __syncthreads();
  for (int idx = t; idx < 64 * 32; idx += 256) {
    int c = idx >> 5, o = idx & 31;
    float s = db[o];
    for (int hh = 0; hh < S4_H; ++hh) s += sH[hh * 64 + c] * dw[hh * 32 + o];
    out[((size_t)(b * S4_L + l0 + c)) * 32 + o] = s;
  }
}

extern "C" void kernel_launch(void* const* d_in, const int* in_sizes, int n_in,
                              void* d_out, int out_size, void* d_ws,
                              size_t ws_size, hipStream_t stream) {
  const float* x = (const float*)d_in[0];
  const float* embed_w = (const float*)d_in[1];
  const float* embed_b = (const float*)d_in[2];
  const float* log_dt = (const float*)d_in[3];
  const float* A_re = (const float*)d_in[4];
  const float* A_im = (const float*)d_in[5];
  const float* C_re = (const float*)d_in[6];
  const float* C_im = (const float*)d_in[7];
  const float* D_skip = (const float*)d_in[8];
  const float* out_w = (const float*)d_in[9];
  const float* out_b = (const float*)d_in[10];
  const float* norm_w = (const float*)d_in[11];
  const float* norm_b = (const float*)d_in[12];
  const float* dec_w = (const float*)d_in[13];
  const float* dec_b = (const float*)d_in[14];

  // workspace layout (floats): ~73.6 MB total
  float* ws = (float*)d_ws;
  const size_t HL = (size_t)S4_B * S4_H * S4_L;  // 8,388,608
  float* hbuf = ws;
  float* ybuf = hbuf + HL;
  float* A1t = ybuf + HL;  // 128*4096
  float* Tt = A1t + 128 * 4096;
  float* Gt = Tt + 128 * 4096;
  float* wq = Gt + 128 * 4096;  // 128*64
  float* Wt = wq + 128 * 64;    // 128*256

  s4_embed_kernel<<<1024, 256, 0, stream>>>(x, embed_w, embed_b, hbuf);
  for (int layer = 0; layer < 4; ++layer) {
    s4_prep_kernel<<<128, 64, 0, stream>>>(
        log_dt + layer * S4_H, A_re + layer * S4_H * S4_N,
        A_im + layer * S4_H * S4_N, C_re + layer * S4_H * S4_N,
        C_im + layer * S4_H * S4_N, out_w + layer * 2 * S4_H * S4_H, A1t, Tt,
        Gt, wq, Wt);
    s4_conv_kernel<<<S4_B * S4_H, 256, 0, stream>>>(hbuf, A1t, Tt, Gt, wq,
                                                    D_skip + layer * S4_H,
                                                    ybuf);
    s4_glu_ln_kernel<<<1024, 256, 0, stream>>>(ybuf, Wt,
                                               out_b + layer * 2 * S4_H, hbuf,
                                               norm_w + layer * S4_H,
                                               norm_b + layer * S4_H);
  }
  s4_decode_kernel<<<1024, 256, 0, stream>>>(hbuf, dec_w, dec_b,
                                             (float*)d_out);
}